// CondMobileNetV2_83476984365405
// MI455X (gfx1250) — compile-verified
//
#include <hip/hip_runtime.h>

// ---------------------------------------------------------------------------
// CondConv-MobileNetV2 forward on MI455X (gfx1250).
// fp32 end-to-end; pointwise convs (85% of FLOPs) run on V_WMMA_F32_16X16X4_F32.
// ---------------------------------------------------------------------------

typedef __attribute__((ext_vector_type(2))) float v2f;
typedef __attribute__((ext_vector_type(8))) float v8f;

#define BATCH 32
#define BN_SLICES 8

// ---------------------------------------------------------------------------
// Stem: 3x3 conv, Cin=3, Cout=32, stride 2, pad 1, 224 -> 112
// ---------------------------------------------------------------------------
__global__ void stem_conv_k(const float* __restrict__ X, const float* __restrict__ W,
                            float* __restrict__ Y) {
    const long total = (long)BATCH * 32 * 112 * 112;
    long i = (long)blockIdx.x * blockDim.x + threadIdx.x;
    if (i >= total) return;
    int ox = (int)(i % 112);
    int oy = (int)((i / 112) % 112);
    int oc = (int)((i / (112 * 112)) % 32);
    int b  = (int)(i / (112L * 112 * 32));
    const float* xb = X + (size_t)b * 3 * 224 * 224;
    const float* w  = W + (size_t)oc * 27;
    float acc = 0.f;
    int iy0 = oy * 2 - 1, ix0 = ox * 2 - 1;
#pragma unroll
    for (int ic = 0; ic < 3; ++ic)
#pragma unroll
        for (int ky = 0; ky < 3; ++ky) {
            int iy = iy0 + ky;
            if ((unsigned)iy >= 224u) continue;
#pragma unroll
            for (int kx = 0; kx < 3; ++kx) {
                int ix = ix0 + kx;
                if ((unsigned)ix >= 224u) continue;
                acc += xb[((size_t)ic * 224 + iy) * 224 + ix] * w[(ic * 3 + ky) * 3 + kx];
            }
        }
    Y[i] = acc;
}

// ---------------------------------------------------------------------------
// BatchNorm (training mode, biased batch var), two-stage reduction for MLP/BW.
// Stage 1: grid (C, BN_SLICES); each block reduces a slice of the batch.
// Stage 2: per-channel finalize -> fused scale/shift:  y = x*scale + shift.
// ---------------------------------------------------------------------------
__global__ void bn_part_k(const float* __restrict__ x, float* __restrict__ part,
                          int C, int HW) {
    int c   = blockIdx.x;
    int sl  = blockIdx.y;
    int tid = threadIdx.x;
    float s1 = 0.f, s2 = 0.f;
    for (int n = sl; n < BATCH; n += BN_SLICES) {
        const float* base = x + ((size_t)n * C + c) * HW;
        for (int p = tid; p < HW; p += blockDim.x) {
            float v = base[p];
            s1 += v; s2 += v * v;
        }
    }
    __shared__ float sh1[256], sh2[256];
    sh1[tid] = s1; sh2[tid] = s2;
    __syncthreads();
    for (int st = 128; st > 0; st >>= 1) {
        if (tid < st) { sh1[tid] += sh1[tid + st]; sh2[tid] += sh2[tid + st]; }
        __syncthreads();
    }
    if (tid == 0) {
        part[((size_t)c * BN_SLICES + sl) * 2]     = sh1[0];
        part[((size_t)c * BN_SLICES + sl) * 2 + 1] = sh2[0];
    }
}

__global__ void bn_fin_k(const float* __restrict__ part, const float* __restrict__ g,
                         const float* __restrict__ b, float* __restrict__ ss,
                         int C, int HW) {
    int c = blockIdx.x * blockDim.x + threadIdx.x;
    if (c >= C) return;
    float s1 = 0.f, s2 = 0.f;
#pragma unroll
    for (int sl = 0; sl < BN_SLICES; ++sl) {
        s1 += part[((size_t)c * BN_SLICES + sl) * 2];
        s2 += part[((size_t)c * BN_SLICES + sl) * 2 + 1];
    }
    float cnt  = (float)BATCH * (float)HW;
    float mean = s1 / cnt;
    float var  = s2 / cnt - mean * mean;
    float sc   = g[c] * rsqrtf(var + 1e-5f);
    ss[2 * c]     = sc;
    ss[2 * c + 1] = b[c] - mean * sc;
}

__global__ void bn_apply_k(float* __restrict__ y, const float* __restrict__ ss,
                           const float* __restrict__ res, int C, int HW,
                           long total, int relu) {
    long i = (long)blockIdx.x * blockDim.x + threadIdx.x;
    if (i >= total) return;
    int c = (int)((i / HW) % C);
    float v = y[i] * ss[2 * c] + ss[2 * c + 1];
    if (res) v += res[i];
    if (relu) v = fminf(fmaxf(v, 0.f), 6.f);
    y[i] = v;
}

// ---------------------------------------------------------------------------
// Depthwise 3x3, pad 1, stride s. wStrideB = per-sample weight stride (0 = shared).
// ---------------------------------------------------------------------------
__global__ void dw_conv_k(const float* __restrict__ X, const float* __restrict__ W,
                          float* __restrict__ Y, int C, int Hin, int Win,
                          int Hout, int Wout, int stride, long wStrideB) {
    long total = (long)BATCH * C * Hout * Wout;
    long i = (long)blockIdx.x * blockDim.x + threadIdx.x;
    if (i >= total) return;
    int ox = (int)(i % Wout);
    int oy = (int)((i / Wout) % Hout);
    int c  = (int)((i / ((long)Wout * Hout)) % C);
    int b  = (int)(i / ((long)Wout * Hout * C));
    const float* xb = X + ((size_t)b * C + c) * Hin * Win;
    const float* w  = W + (size_t)b * wStrideB + (size_t)c * 9;
    float acc = 0.f;
    int iy0 = oy * stride - 1, ix0 = ox * stride - 1;
#pragma unroll
    for (int ky = 0; ky < 3; ++ky) {
        int iy = iy0 + ky;
        if ((unsigned)iy >= (unsigned)Hin) continue;
#pragma unroll
        for (int kx = 0; kx < 3; ++kx) {
            int ix = ix0 + kx;
            if ((unsigned)ix >= (unsigned)Win) continue;
            acc += xb[(size_t)iy * Win + ix] * w[ky * 3 + kx];
        }
    }
    Y[i] = acc;
}

// ---------------------------------------------------------------------------
// Pointwise conv as WMMA GEMM:  Y[b][Cout][HW] = W_b[Cout][Cin] * X[b][Cin][HW]
// One wave computes a 16(M) x 64(N) tile, K stepped by 4 via v_wmma_f32_16x16x4_f32.
// A-fragment (16x4 fp32, ISA layout): lane<16 holds row m0+lane K={0,1};
// lane>=16 same row K={2,3} -> one aligned v2f (b64) load per K-step.
// B-fragment (4x16): lanes 0-15 col n, vgpr=K{0,1}; lanes 16-31 K{2,3}.
// C/D: vgpr v -> row m0 + v + 8*(lane>>4), col n0 + (lane&15).
// All Cin in this net are multiples of 4 -> no K guard needed.
// ---------------------------------------------------------------------------
__global__ void pw_gemm_k(const float* __restrict__ W, const float* __restrict__ X,
                          float* __restrict__ Y, int Cout, int Cin, int HW,
                          int Mt, int Nt, long wStrideB) {
    int b  = blockIdx.z;
    int gw = blockIdx.x * (blockDim.x >> 5) + (threadIdx.x >> 5);
    if (gw >= Mt * Nt) return;              // uniform per wave: EXEC stays all-ones
    int mtile = gw % Mt;
    int ntile = gw / Mt;
    int lane = threadIdx.x & 31;
    int lr   = lane & 15;
    int kb   = (lane >> 4) * 2;

    const float* Wb = W + (size_t)b * wStrideB;
    const float* Xb = X + (size_t)b * Cin * HW;
    float*       Yb = Y + (size_t)b * Cout * HW;

    int m0 = mtile * 16, n0 = ntile * 64;
    int row  = m0 + lr;
    int rowc = row < Cout ? row : Cout - 1;   // clamp (guarded at store)
    const float* wrow = Wb + (size_t)rowc * Cin + kb;  // 8B-aligned (Cin%4==0)

    int col[4];
    const float* xc[4];
#pragma unroll
    for (int j = 0; j < 4; ++j) {
        int c = n0 + j * 16 + lr;
        col[j] = c;
        xc[j]  = Xb + (c < HW ? c : 0);       // clamp (guarded at store)
    }

    v8f acc0 = {}, acc1 = {}, acc2 = {}, acc3 = {};
    for (int k = 0; k < Cin; k += 4) {
        v2f a = *(const v2f*)(wrow + k);      // single global_load_b64
        size_t o0 = (size_t)(k + kb) * HW;
        size_t o1 = o0 + HW;
        v2f b0, b1, b2, b3;
        b0.x = xc[0][o0]; b0.y = xc[0][o1];
        b1.x = xc[1][o0]; b1.y = xc[1][o1];
        b2.x = xc[2][o0]; b2.y = xc[2][o1];
        b3.x = xc[3][o0]; b3.y = xc[3][o1];
        acc0 = __builtin_amdgcn_wmma_f32_16x16x4_f32(false, a, false, b0, (short)0, acc0, false, false);
        acc1 = __builtin_amdgcn_wmma_f32_16x16x4_f32(false, a, false, b1, (short)0, acc1, false, false);
        acc2 = __builtin_amdgcn_wmma_f32_16x16x4_f32(false, a, false, b2, (short)0, acc2, false, false);
        acc3 = __builtin_amdgcn_wmma_f32_16x16x4_f32(false, a, false, b3, (short)0, acc3, false, false);
    }

    int rbase = m0 + (lane >> 4) * 8;
    // Store straight out of the accumulators (no v8f array copy -> no mov block).
#define PW_STORE(J, ACC)                                                     \
    {                                                                        \
        int c = col[J];                                                      \
        if (c < HW) {                                                        \
            _Pragma("unroll") for (int v = 0; v < 8; ++v) {                  \
                int r = rbase + v;                                           \
                if (r < Cout) Yb[(size_t)r * HW + c] = ACC[v];               \
            }                                                                \
        }                                                                    \
    }
    PW_STORE(0, acc0)
    PW_STORE(1, acc1)
    PW_STORE(2, acc2)
    PW_STORE(3, acc3)
#undef PW_STORE
}

// ---------------------------------------------------------------------------
// Global average pool over HW: P[b][c]
// ---------------------------------------------------------------------------
__global__ void pool_k(const float* __restrict__ X, float* __restrict__ P,
                       int C, int HW) {
    int i = blockIdx.x * blockDim.x + threadIdx.x;
    if (i >= BATCH * C) return;
    int b = i / C, c = i % C;
    const float* xb = X + ((size_t)b * C + c) * HW;
    float s = 0.f;
    for (int p = 0; p < HW; ++p) s += xb[p];
    P[i] = s / (float)HW;
}

// ---------------------------------------------------------------------------
// Routing: R[b][k] = sigmoid(P[b]·RW[k] + RB[k])
// ---------------------------------------------------------------------------
__global__ void route_k(const float* __restrict__ P, const float* __restrict__ RW,
                        const float* __restrict__ RB, float* __restrict__ R,
                        int C, int K) {
    int i = blockIdx.x * blockDim.x + threadIdx.x;
    if (i >= BATCH * K) return;
    int b = i / K, k = i % K;
    const float* p = P + (size_t)b * C;
    const float* w = RW + (size_t)k * C;
    float z = RB[k];
    for (int c = 0; c < C; ++c) z += p[c] * w[c];
    R[i] = 1.f / (1.f + expf(-z));
}

// ---------------------------------------------------------------------------
// Expert weight mixing: CW[b][e] = sum_k R[b][k] * W[k][e]
// ---------------------------------------------------------------------------
__global__ void mix_k(const float* __restrict__ W, const float* __restrict__ R,
                      float* __restrict__ CW, long wsz, int K) {
    long total = (long)BATCH * wsz;
    long i = (long)blockIdx.x * blockDim.x + threadIdx.x;
    if (i >= total) return;
    int  b = (int)(i / wsz);
    long e = i % wsz;
    float s = 0.f;
    for (int k = 0; k < K; ++k) s += R[b * K + k] * W[(size_t)k * wsz + e];
    CW[i] = s;
}

// ---------------------------------------------------------------------------
// CondConv classifier: logits[b][o] = sum_k R[b][k] * (P[b] · cls_w[k][o][:])
// ---------------------------------------------------------------------------
__global__ void cls_k(const float* __restrict__ P, const float* __restrict__ R,
                      const float* __restrict__ W, float* __restrict__ Out) {
    int i = blockIdx.x * blockDim.x + threadIdx.x;
    if (i >= BATCH * 1000) return;
    int b = i / 1000, o = i % 1000;
    const float* pb = P + (size_t)b * 1280;
    float acc = 0.f;
    for (int k = 0; k < 8; ++k) {
        const float* w = W + ((size_t)k * 1000 + o) * 1280;
        float d = 0.f;
        for (int c = 0; c < 1280; ++c) d += pb[c] * w[c];
        acc += R[b * 8 + k] * d;
    }
    Out[i] = acc;
}

// ---------------------------------------------------------------------------
// Host orchestration
// ---------------------------------------------------------------------------
struct Spec { int inp, oup, s, t, ne; };
static const Spec SPECS[17] = {
    {32, 16, 1, 1, 0}, {16, 24, 2, 6, 0}, {24, 24, 1, 6, 0},
    {24, 32, 2, 6, 0}, {32, 32, 1, 6, 0}, {32, 32, 1, 6, 0},
    {32, 64, 2, 6, 0}, {64, 64, 1, 6, 0}, {64, 64, 1, 6, 0}, {64, 64, 1, 6, 0},
    {64, 96, 1, 6, 0}, {96, 96, 1, 6, 8}, {96, 96, 1, 6, 8},
    {96, 160, 2, 6, 8}, {160, 160, 1, 6, 8}, {160, 160, 1, 6, 8},
    {160, 320, 1, 6, 8}
};

struct BP {
    const float *dw_b, *dw_g, *dw_w, *pl_b, *pl_g, *pl_w;
    const float *pw_b, *pw_g, *pw_w, *route_b, *route_w;
};

extern "C" void kernel_launch(void* const* d_in, const int* in_sizes, int n_in,
                              void* d_out, int out_size, void* d_ws, size_t ws_size,
                              hipStream_t stream) {
    (void)out_size; (void)ws_size;

    // ---- locate x (unique element count) and walk params in sorted-key
    //      pytree-flatten order: blocks[*]{dw_b,dw_g,dw_w,pl_b,pl_g,pl_w,
    //      pw_b,pw_g,pw_w,route_b,route_w}, cls_route_b, cls_route_w, cls_w,
    //      head_b, head_g, head_w, stem_b, stem_g, stem_w.
    const long X_ELEMS = (long)BATCH * 3 * 224 * 224;
    const float* x_in;
    int pi;
    if (n_in > 0 && (long)in_sizes[0] == X_ELEMS) { x_in = (const float*)d_in[0]; pi = 1; }
    else { x_in = (const float*)d_in[n_in - 1]; pi = 0; }
    auto nxt = [&]() { return (const float*)d_in[pi++]; };

    BP bp[17];
    for (int i = 0; i < 17; ++i) {
        bp[i].dw_b = nxt(); bp[i].dw_g = nxt(); bp[i].dw_w = nxt();
        bp[i].pl_b = nxt(); bp[i].pl_g = nxt(); bp[i].pl_w = nxt();
        if (SPECS[i].t != 1) { bp[i].pw_b = nxt(); bp[i].pw_g = nxt(); bp[i].pw_w = nxt(); }
        else { bp[i].pw_b = bp[i].pw_g = bp[i].pw_w = nullptr; }
        if (SPECS[i].ne) { bp[i].route_b = nxt(); bp[i].route_w = nxt(); }
        else { bp[i].route_b = bp[i].route_w = nullptr; }
    }
    const float* cls_route_b = nxt();
    const float* cls_route_w = nxt();
    const float* cls_w       = nxt();
    const float* head_b = nxt(); const float* head_g = nxt(); const float* head_w = nxt();
    const float* stem_b = nxt(); const float* stem_g = nxt(); const float* stem_w = nxt();

    // ---- workspace carve (floats). Largest activation: 32*96*112*112.
    float* ws = (float*)d_ws;
    const size_t BUFSZ = (size_t)BATCH * 96 * 112 * 112;
    float* B0     = ws;
    float* B1     = ws + BUFSZ;
    float* B2     = ws + 2 * BUFSZ;
    float* stats  = ws + 3 * BUFSZ;                    // 2*1280
    float* part   = stats + 2 * 1280;                  // 1280*BN_SLICES*2
    float* pooled = part + (size_t)1280 * BN_SLICES * 2;  // 32*1280
    float* rwbuf  = pooled + (size_t)BATCH * 1280;     // 32*8
    float* mixW   = rwbuf + (size_t)BATCH * 8;         // up to 32*320*960

    auto nblk = [](long total) { return (unsigned)((total + 255) / 256); };

    auto run_bn = [&](float* x, const float* g, const float* b, int C, int HW,
                      const float* res, int relu) {
        bn_part_k<<<dim3((unsigned)C, BN_SLICES), 256, 0, stream>>>(x, part, C, HW);
        bn_fin_k<<<(C + 255) / 256, 256, 0, stream>>>(part, g, b, stats, C, HW);
        long total = (long)BATCH * C * HW;
        bn_apply_k<<<nblk(total), 256, 0, stream>>>(x, stats, res, C, HW, total, relu);
    };
    auto run_pw = [&](const float* W, const float* X, float* Y, int Cout, int Cin,
                      int HW, long wStrideB) {
        int Mt = (Cout + 15) / 16;
        int Nt = (HW + 63) / 64;
        int waves = Mt * Nt;
        dim3 grid((unsigned)((waves + 7) / 8), 1, BATCH);
        pw_gemm_k<<<grid, 256, 0, stream>>>(W, X, Y, Cout, Cin, HW, Mt, Nt, wStrideB);
    };
    auto run_mix = [&](const float* W, long wsz) {
        long total = (long)BATCH * wsz;
        mix_k<<<nblk(total), 256, 0, stream>>>(W, rwbuf, mixW, wsz, 8);
    };

    // ---- stem: conv 3x3 s2 + BN + relu6 ----
    {
        long total = (long)BATCH * 32 * 112 * 112;
        stem_conv_k<<<nblk(total), 256, 0, stream>>>(x_in, stem_w, B0);
        run_bn(B0, stem_g, stem_b, 32, 112 * 112, nullptr, 1);
    }

    float* X  = B0;
    float* T1 = B1;
    float* T2 = B2;
    int H = 112;

    for (int i = 0; i < 17; ++i) {
        const Spec sp = SPECS[i];
        const BP&  p  = bp[i];
        int hid = sp.inp * sp.t;
        int HW  = H * H;
        int Ho  = (H + 2 - 3) / sp.s + 1;
        int HWo = Ho * Ho;

        if (sp.ne) {  // routing from block input
            pool_k<<<nblk((long)BATCH * sp.inp), 256, 0, stream>>>(X, pooled, sp.inp, HW);
            route_k<<<1, 256, 0, stream>>>(pooled, p.route_w, p.route_b, rwbuf, sp.inp, 8);
        }

        const float* dwsrc = X;
        if (sp.t != 1) {  // expand pointwise
            const float* Wp = p.pw_w;
            long strideB = 0;
            if (sp.ne) { run_mix(p.pw_w, (long)hid * sp.inp); Wp = mixW; strideB = (long)hid * sp.inp; }
            run_pw(Wp, X, T1, hid, sp.inp, HW, strideB);
            run_bn(T1, p.pw_g, p.pw_b, hid, HW, nullptr, 1);
            dwsrc = T1;
        }

        {  // depthwise
            const float* Wd = p.dw_w;
            long strideB = 0;
            if (sp.ne) { run_mix(p.dw_w, (long)hid * 9); Wd = mixW; strideB = (long)hid * 9; }
            long total = (long)BATCH * hid * HWo;
            dw_conv_k<<<nblk(total), 256, 0, stream>>>(dwsrc, Wd, T2, hid, H, H, Ho, Ho, sp.s, strideB);
            run_bn(T2, p.dw_g, p.dw_b, hid, HWo, nullptr, 1);
        }

        {  // project pointwise (+ residual in BN apply)
            const float* Wl = p.pl_w;
            long strideB = 0;
            if (sp.ne) { run_mix(p.pl_w, (long)sp.oup * hid); Wl = mixW; strideB = (long)sp.oup * hid; }
            run_pw(Wl, T2, T1, sp.oup, hid, HWo, strideB);
            bool resid = (sp.s == 1 && sp.inp == sp.oup);
            run_bn(T1, p.pl_g, p.pl_b, sp.oup, HWo, resid ? X : nullptr, 0);
        }

        float* oldX = X;
        X  = T1;      // block output
        T1 = oldX;    // recycle
        H  = Ho;
    }

    // ---- head: 1x1 conv 320->1280 + BN + relu6, pool, route, classifier ----
    run_pw(head_w, X, T1, 1280, 320, 49, 0);
    run_bn(T1, head_g, head_b, 1280, 49, nullptr, 1);
    pool_k<<<nblk((long)BATCH * 1280), 256, 0, stream>>>(T1, pooled, 1280, 49);
    route_k<<<1, 256, 0, stream>>>(pooled, cls_route_w, cls_route_b, rwbuf, 1280, 8);
    cls_k<<<nblk((long)BATCH * 1000), 256, 0, stream>>>(pooled, rwbuf, cls_w, (float*)d_out);
}